// WaveletNet_86638080295338
// MI455X (gfx1250) — compile-verified
//
#include <hip/hip_runtime.h>
#include <cstdint>

#define BDIM 256
#define PAIRS_PER_THREAD 4
#define PAIRS_PER_BLOCK (BDIM * PAIRS_PER_THREAD)   // 1024 pairs / block
#define LROW 65536                                   // floats per input row
#define HALF (LROW / 2)                              // 32768 pairs per row
#define NROWS 256
#define NSTEPS 4

// Native clang vector (HIP's float4 is a struct and is rejected by
// __builtin_nontemporal_store).
typedef float v4f __attribute__((ext_vector_type(4)));

// Lifting step k maps [e;o] -> [[1-u*p, u], [-p, 1]] [e;o].  The whole
// transform is the product M = M3*M2*M1*M0 applied pointwise to each
// (even, odd) pair: avg = m00*e + m01*o, det = m10*e + m11*o.
__global__ __launch_bounds__(BDIM) void wavelet_lift_kernel(
    const float* __restrict__ x,
    const float* __restrict__ P,
    const float* __restrict__ U,
    float* __restrict__ out)
{
    __shared__ float tile[PAIRS_PER_BLOCK * 2];      // 8 KiB staging buffer

    const int tid     = threadIdx.x;
    const int row     = (int)(blockIdx.x >> 5);      // 32 blocks per row
    const int tileIdx = (int)(blockIdx.x & 31);
    const int pair0   = tileIdx * PAIRS_PER_BLOCK + tid * PAIRS_PER_THREAD;

    // This lane's 32 contiguous input bytes (two interleaved (e,o) float4s).
    const float* gsrc = x + (size_t)row * LROW + (size_t)pair0 * 2;

    // Low 32 bits of the generic shared-memory address == LDS byte offset
    // (flat->LDS addresses truncate to addr[31:0] per the CDNA5 aperture rule).
    float* slot = &tile[tid * (PAIRS_PER_THREAD * 2)];
    unsigned lds_off = (unsigned)(uintptr_t)slot;

    // CDNA5 async data-mover path: stream global -> LDS, tracked by ASYNCcnt.
    // The inst offset applies to both the LDS and global address, so one base
    // covers both b128 transfers.  Each lane reads back only its own bytes,
    // so a wave-local s_wait_asynccnt is sufficient (no workgroup barrier).
    asm volatile(
        "global_load_async_to_lds_b128 %0, %1, off\n\t"
        "global_load_async_to_lds_b128 %0, %1, off offset:16\n\t"
        "s_wait_asynccnt 0"
        :
        : "v"(lds_off), "v"(gsrc)
        : "memory");

    // Compose the 4 lifting steps into one 2x2 matrix (register-only, ~16 FMA).
    float m00 = 1.0f, m01 = 0.0f, m10 = 0.0f, m11 = 1.0f;
#pragma unroll
    for (int k = 0; k < NSTEPS; ++k) {
        const float p = P[k];
        const float u = U[k];
        const float a00 = fmaf(-u, p, 1.0f);         // 1 - u*p
        const float n00 = fmaf(a00, m00, u * m10);
        const float n01 = fmaf(a00, m01, u * m11);
        const float n10 = fmaf(-p, m00, m10);
        const float n11 = fmaf(-p, m01, m11);
        m00 = n00; m01 = n01; m10 = n10; m11 = n11;
    }

    // Read staged pairs back from LDS (ds_load_b128 x2).
    const v4f* t4 = (const v4f*)slot;
    const v4f i0 = t4[0];                            // e0 o0 e1 o1
    const v4f i1 = t4[1];                            // e2 o2 e3 o3

    v4f avg, det;
    avg.x = fmaf(m00, i0.x, m01 * i0.y);  det.x = fmaf(m10, i0.x, m11 * i0.y);
    avg.y = fmaf(m00, i0.z, m01 * i0.w);  det.y = fmaf(m10, i0.z, m11 * i0.w);
    avg.z = fmaf(m00, i1.x, m01 * i1.y);  det.z = fmaf(m10, i1.x, m11 * i1.y);
    avg.w = fmaf(m00, i1.z, m01 * i1.w);  det.w = fmaf(m10, i1.z, m11 * i1.w);

    // z layout: out[row*65536 + half*32768 + i]; both halves stay 16B aligned.
    // Non-temporal stores: output is written once and never re-read, so keep
    // the streamed input resident in the 192MB L2 across graph replays.
    float* orow = out + (size_t)row * LROW + (size_t)pair0;
    __builtin_nontemporal_store(avg, (v4f*)orow);
    __builtin_nontemporal_store(det, (v4f*)(orow + HALF));

    // log_det = 0.0f, appended after the flattened z.
    if (blockIdx.x == 0 && tid == 0) {
        out[(size_t)NROWS * LROW] = 0.0f;
    }
}

extern "C" void kernel_launch(void* const* d_in, const int* in_sizes, int n_in,
                              void* d_out, int out_size, void* d_ws, size_t ws_size,
                              hipStream_t stream) {
    (void)in_sizes; (void)n_in; (void)out_size; (void)d_ws; (void)ws_size;
    const float* x = (const float*)d_in[0];   // (256, 65536) f32
    const float* P = (const float*)d_in[1];   // (4, 1) f32
    const float* U = (const float*)d_in[2];   // (4, 1) f32
    float* out = (float*)d_out;               // 256*2*32768 z values + 1 log_det

    const int blocks = NROWS * (HALF / PAIRS_PER_BLOCK);   // 256 * 32 = 8192
    wavelet_lift_kernel<<<blocks, BDIM, 0, stream>>>(x, P, U, out);
}